// LogSimpleSlater_75411035783169
// MI455X (gfx1250) — compile-verified
//
#include <hip/hip_runtime.h>
#include <hip/hip_bf16.h>

// Problem constants from the reference.
#define N_ELEC   128
#define DIM      3
#define LDS_LD   129   // padded leading dim (floats): 129 mod 64(banks) = 1 -> conflict-free columns
#define NTHREADS 256   // 8 waves (wave32)
#define NB       16    // LU block size (matches WMMA tile)

typedef __attribute__((ext_vector_type(2))) float v2f;
typedef __attribute__((ext_vector_type(8))) float v8f;

// One workgroup per walker.
//  Phase 1: 128x128 Slater matrix in LDS via V_WMMA_F32_16X16X4_F32 (K=3 padded to 4).
//  Phase 2: blocked right-looking LU (NB=16) with partial pivoting; the trailing
//           submatrix update C -= L_panel * U_row runs on WMMA (4 chained K=4 ops
//           per 16x16x16 tile), which is where >80% of the LU FLOPs are.
//  Phase 3: log|det| = sum log|U_ii| reduction.
__global__ __launch_bounds__(NTHREADS)
void slater_logdet_kernel(const float* __restrict__ rs,       // [B, 128, 3]
                          const float* __restrict__ kpoints,  // [128, 3]
                          const float* __restrict__ cos_sw,   // [128]
                          const float* __restrict__ sin_sw,   // [128]
                          float* __restrict__ out)            // [B]
{
    extern __shared__ float smem[];
    float* A       = smem;                       // 128 * 129 floats
    float* red_val = smem + N_ELEC * LDS_LD;     // 128 floats
    int*   red_idx = (int*)(red_val + N_ELEC);   // 128 ints

    const int b    = blockIdx.x;
    const int tid  = threadIdx.x;
    const int wave = tid >> 5;    // 0..7
    const int lane = tid & 31;

    // WMMA fragment index helpers (wave32, f32 shapes):
    //  A-frag 16x4 : lane -> M = lane&15, K = 2*(lane>>4) + {0,1}
    //  B-frag 4x16 : lane -> N = lane&15, K = 2*(lane>>4) + {0,1}
    //  C/D  16x16  : (vgpr r, lane) -> M = r + 8*(lane>>4), N = lane&15
    const int mrow  = lane & 15;
    const int kpair = (lane >> 4) * 2;            // 0 or 2
    const int ncol  = lane & 15;
    const int rsel  = lane >> 4;                  // 0: M=r, 1: M=r+8
    // Branchless K=3 zero-padding: load from a clamped in-bounds address, then
    // select 0 via v_cndmask (avoids divergent-EXEC branches around the load).
    const float padmask = (kpair == 0) ? 1.0f : 0.0f;   // second K lane valid only for kpair=0
    const int   MAXIDX  = N_ELEC * DIM - 1;             // 383

    // ---------------- Phase 1: Slater matrix via WMMA ----------------
    const float* rsb = rs + (size_t)b * N_ELEC * DIM;

    v2f afrag;
    {
        const int m    = 16 * wave + mrow;        // electron row this lane feeds
        const int i0   = m * DIM + kpair;         // k in {0,2}: always valid
        const int i1   = min(i0 + 1, MAXIDX);     // clamped, always in-bounds
        afrag.x = rsb[i0];
        afrag.y = rsb[i1] * padmask;              // k=3 padded to 0
    }

    for (int nt = 0; nt < 8; ++nt) {
        const int n  = 16 * nt + mrow;
        const int i0 = n * DIM + kpair;
        const int i1 = min(i0 + 1, MAXIDX);
        v2f bfrag;
        bfrag.x = kpoints[i0];
        bfrag.y = kpoints[i1] * padmask;

        v8f c = {};
        c = __builtin_amdgcn_wmma_f32_16x16x4_f32(false, afrag, false, bfrag,
                                                  (short)0, c, false, false);

        const int   N  = 16 * nt + ncol;
        const float cw = cos_sw[N];
        const float sw = sin_sw[N];
#pragma unroll
        for (int r = 0; r < 8; ++r) {
            const int   M = 16 * wave + r + 8 * rsel;
            const float d = c[r];
            // cos_sw*cos(d) + sin_sw*sin(-d)
            A[M * LDS_LD + N] = cw * __cosf(d) - sw * __sinf(d);
        }
    }
    __syncthreads();

    // ---------------- Phase 2: blocked LU with partial pivoting ----------------
    for (int kb = 0; kb < N_ELEC; kb += NB) {
        // ---- Panel factorization: tall panel rows kb..127, cols kb..kb+15 ----
        for (int k = kb; k < kb + NB; ++k) {
            // Parallel argmax of |A[i][k]| over i >= k.
            if (tid < N_ELEC) {
                red_val[tid] = (tid >= k) ? fabsf(A[tid * LDS_LD + k]) : -1.0f;
                red_idx[tid] = tid;
            }
            __syncthreads();
            for (int s = 64; s > 0; s >>= 1) {
                if (tid < s) {
                    if (red_val[tid + s] > red_val[tid]) {
                        red_val[tid] = red_val[tid + s];
                        red_idx[tid] = red_idx[tid + s];
                    }
                }
                __syncthreads();
            }
            const int p = red_idx[0];

            // Swap full rows k <-> p (keeps L history consistent).
            if (p != k) {
                for (int j = tid; j < N_ELEC; j += NTHREADS) {
                    const float t     = A[k * LDS_LD + j];
                    A[k * LDS_LD + j] = A[p * LDS_LD + j];
                    A[p * LDS_LD + j] = t;
                }
            }
            __syncthreads();

            const float piv = A[k * LDS_LD + k];

            // Rank-1 update restricted to the panel columns (<= 15 wide).
            const int i = k + 1 + tid;
            if (i < N_ELEC) {
                const float l = A[i * LDS_LD + k] / piv;
                A[i * LDS_LD + k] = l;
                for (int j = k + 1; j < kb + NB; ++j)
                    A[i * LDS_LD + j] -= l * A[k * LDS_LD + j];
            }
            __syncthreads();
        }

        const int t0 = kb + NB;           // start of trailing submatrix
        if (t0 >= N_ELEC) break;

        // ---- Triangular solve: U(kb:kb+16, t0:128) = L_kk^-1 * A(kb:kb+16, t0:128)
        // Column-per-thread; L entries are uniform LDS broadcasts.
        {
            const int j = t0 + tid;
            if (j < N_ELEC) {
                float v[NB];
#pragma unroll
                for (int r = 0; r < NB; ++r) v[r] = A[(kb + r) * LDS_LD + j];
#pragma unroll
                for (int r = 1; r < NB; ++r) {
#pragma unroll
                    for (int s = 0; s < NB; ++s) {
                        if (s < r) v[r] -= A[(kb + r) * LDS_LD + (kb + s)] * v[s];
                    }
                }
#pragma unroll
                for (int r = 0; r < NB; ++r) A[(kb + r) * LDS_LD + j] = v[r];
            }
        }
        __syncthreads();

        // ---- Trailing update: A(t0:,t0:) -= L(t0:,kb:kb+16) * U(kb:kb+16,t0:)
        // 16x16 tiles round-robin over the 8 waves; 4 chained K=4 WMMAs per tile.
        {
            const int ntile = (N_ELEC - t0) >> 4;      // tiles per side: 7..1
            for (int tile = wave; tile < ntile * ntile; tile += 8) {
                const int ti = tile / ntile;
                const int tj = tile - ti * ntile;
                const int i0 = t0 + 16 * ti;
                const int j0 = t0 + 16 * tj;

                v8f c = {};
#pragma unroll
                for (int kk = 0; kk < 4; ++kk) {       // K chunks of 4 over the panel
                    const int k0 = kb + 4 * kk + kpair;
                    v2f af, bf;
                    af.x = A[(i0 + mrow) * LDS_LD + k0];
                    af.y = A[(i0 + mrow) * LDS_LD + k0 + 1];
                    bf.x = A[(k0)     * LDS_LD + j0 + ncol];
                    bf.y = A[(k0 + 1) * LDS_LD + j0 + ncol];
                    c = __builtin_amdgcn_wmma_f32_16x16x4_f32(false, af, false, bf,
                                                              (short)0, c, false, false);
                }

#pragma unroll
                for (int r = 0; r < 8; ++r) {
                    const int M = i0 + r + 8 * rsel;
                    const int N = j0 + ncol;
                    A[M * LDS_LD + N] -= c[r];
                }
            }
        }
        __syncthreads();
    }

    // ---------------- Phase 3: log|det| = sum log|U_ii| ----------------
    if (tid < N_ELEC)
        red_val[tid] = __logf(fabsf(A[tid * LDS_LD + tid]));
    __syncthreads();
    for (int s = 64; s > 0; s >>= 1) {
        if (tid < s) red_val[tid] += red_val[tid + s];
        __syncthreads();
    }
    if (tid == 0) out[b] = red_val[0];
}

extern "C" void kernel_launch(void* const* d_in, const int* in_sizes, int n_in,
                              void* d_out, int out_size, void* d_ws, size_t ws_size,
                              hipStream_t stream)
{
    const float* rs      = (const float*)d_in[0];  // [B,128,3]
    const float* kpoints = (const float*)d_in[1];  // [128,3]
    const float* cos_sw  = (const float*)d_in[2];  // [128]
    const float* sin_sw  = (const float*)d_in[3];  // [128]
    float*       out     = (float*)d_out;          // [B]

    const int B = in_sizes[0] / (N_ELEC * DIM);
    const size_t shmem = (size_t)(N_ELEC * LDS_LD + N_ELEC) * sizeof(float)
                       + (size_t)N_ELEC * sizeof(int);

    slater_logdet_kernel<<<B, NTHREADS, shmem, stream>>>(rs, kpoints, cos_sw, sin_sw, out);
}